// CodebookEMA_41592463294506
// MI455X (gfx1250) — compile-verified
//
#include <hip/hip_runtime.h>
#include <math.h>

#define K_CODES 16384
#define DIMS    128
#define BATCH_N 262144
#define DECAY_F 0.99f
#define EPS_F   1e-5f

typedef __attribute__((ext_vector_type(2))) float v2f;
typedef __attribute__((ext_vector_type(8))) float v8f;

// Kernel 1: zero the w accumulator (aliased onto d_out) and the per-code counts.
__global__ void ema_init_kernel(float* __restrict__ w, float* __restrict__ cnt) {
  const int stride = gridDim.x * blockDim.x;
  const int i = blockIdx.x * blockDim.x + threadIdx.x;
  const int total = K_CODES * DIMS;
  for (int j = i; j < total; j += stride) w[j] = 0.0f;
  for (int j = i; j < K_CODES; j += stride) cnt[j] = 0.0f;
}

// Kernel 2: segment-sum scatter. Each 256-thread block owns a 16-row batch tile.
// Wave wv (0..7) owns columns [16*wv, 16*wv+16). Intra-tile duplicate codes are
// merged via D = M @ Z with chained V_WMMA_F32_16X16X4_F32 (M = exact 0/1
// leader-gather matrix). Non-leader rows of D are exactly zero and target the
// same code as their leader, so the scatter atomics run unconditionally.
// Lane-varying code gathers use ds_bpermute (cheap LDS-pipe ops, no select
// trees); all global offsets are 32-bit so loads/atomics use saddr+voffset.
__global__ __launch_bounds__(256)
void ema_scatter_kernel(const float* __restrict__ z_e, const int* __restrict__ idxs,
                        float* __restrict__ w, float* __restrict__ cnt) {
  const int lane = threadIdx.x & 31;
  const int wv   = threadIdx.x >> 5;       // wave in block: 0..7
  const int r    = lane & 15;              // row (A/M) or column (B/D) index 0..15
  const int half = lane >> 4;              // 0 = lanes 0-15, 1 = lanes 16-31
  const unsigned base = blockIdx.x * 16u;  // first batch row of this tile
  const int colbase = wv * 16;             // first feature column of this wave

  // Each lane holds the code index of tile-row r (duplicated across halves).
  const int rowIdx = idxs[base + (unsigned)r];

  // matchmask bit c: tile-row c carries the same code as this lane's row.
  unsigned mm = 0u;
  #pragma unroll
  for (int c = 0; c < 16; ++c) {
    mm |= (unsigned)(__shfl(rowIdx, c, 32) == rowIdx) << c;
  }

  // Leader = first tile-row carrying this code; non-leaders get a zero A row.
  const bool isLeader = ((mm & ((1u << r) - 1u)) == 0u);
  const unsigned gmask = isLeader ? mm : 0u;

  // D(16x16) = sum_k M_k(16x4) @ Z_k(4x16), chained f32 WMMA, K split 4x4.
  // A layout: lanes 0-15 hold K={4k,4k+1}, lanes 16-31 hold K={4k+2,4k+3}.
  // B layout mirrors A's K striping; N = lane&15.
  v8f acc = {};
  #pragma unroll
  for (int k = 0; k < 4; ++k) {
    const int j0 = 4 * k + 2 * half;
    v2f a, b;
    a.x = (float)((gmask >> j0) & 1u);
    a.y = (float)((gmask >> (j0 + 1)) & 1u);
    const unsigned zoff = (base + (unsigned)j0) * DIMS + (unsigned)(colbase + r);
    b.x = z_e[zoff];
    b.y = z_e[zoff + DIMS];
    acc = __builtin_amdgcn_wmma_f32_16x16x4_f32(false, a, false, b,
                                                (short)0, acc, false, false);
  }

  // D layout: VGPR v -> tile-row (8*half + v), column (lane&15).
  // Target code via one ds_bpermute per row; 32-bit offsets; unconditional.
  #pragma unroll
  for (int v = 0; v < 8; ++v) {
    const int tcode = __shfl(rowIdx, 8 * half + v, 32);
    const unsigned off = (unsigned)tcode * DIMS + (unsigned)(colbase + r);
    atomicAdd(&w[off], acc[v]);
  }

  // Counts: one atomic per distinct code in the tile (wave 0, lanes 0-15 only).
  if (wv == 0 && half == 0 && isLeader) {
    atomicAdd(&cnt[(unsigned)rowIdx], (float)__popc(mm));
  }
}

// Kernel 3: EMA update + debias + divide, in place over the w accumulator.
__global__ void ema_finalize_kernel(const float* __restrict__ m, const float* __restrict__ N,
                                    const float* __restrict__ cnt, const int* __restrict__ tptr,
                                    float* __restrict__ out) {
  const float zbc = 1.0f - powf(DECAY_F, (float)tptr[0]);
  const int stride = gridDim.x * blockDim.x;
  const int total = K_CODES * DIMS;
  for (int j = blockIdx.x * blockDim.x + threadIdx.x; j < total; j += stride) {
    const int k = j >> 7;  // j / DIMS
    const float wv = out[j];
    const float nU = N[k] * DECAY_F + cnt[k] * (1.0f - DECAY_F);
    const float mU = m[j] * DECAY_F + wv * (1.0f - DECAY_F);
    out[j] = (mU / zbc) / ((nU / zbc) + EPS_F);
  }
}

extern "C" void kernel_launch(void* const* d_in, const int* in_sizes, int n_in,
                              void* d_out, int out_size, void* d_ws, size_t ws_size,
                              hipStream_t stream) {
  const float* z_e  = (const float*)d_in[0];   // (BATCH, D)
  const int*   idxs = (const int*)  d_in[1];   // (BATCH,)
  const float* m    = (const float*)d_in[2];   // (K, D)
  const float* N    = (const float*)d_in[3];   // (K, 1)
  const int*   t    = (const int*)  d_in[4];   // scalar
  float* out = (float*)d_out;                  // w accumulator, then e (K, D)
  float* cnt = (float*)d_ws;                   // K floats of scratch

  ema_init_kernel<<<2048, 256, 0, stream>>>(out, cnt);
  ema_scatter_kernel<<<BATCH_N / 16, 256, 0, stream>>>(z_e, idxs, out, cnt);
  ema_finalize_kernel<<<(K_CODES * DIMS) / 256, 256, 0, stream>>>(m, N, cnt, t, out);
  (void)in_sizes; (void)n_in; (void)out_size; (void)ws_size;
}